// TileVoxelizer_85718957293931
// MI455X (gfx1250) — compile-verified
//
#include <hip/hip_runtime.h>

#define D_DIM 256
#define H_DIM 256
#define W_DIM 256
#define RWIN  6

typedef __attribute__((ext_vector_type(2))) float v2f;
typedef __attribute__((ext_vector_type(8))) float v8f;

__device__ __forceinline__ void atomicAddF32(float* p, float v) {
    // relaxed, device scope -> should lower to global_atomic_add_f32 (no CAS loop)
    __hip_atomic_fetch_add(p, v, __ATOMIC_RELAXED, __HIP_MEMORY_SCOPE_AGENT);
}

__global__ void tv_zero_kernel(float4* __restrict__ out, int n4) {
    int i = blockIdx.x * blockDim.x + threadIdx.x;
    if (i < n4) out[i] = make_float4(0.f, 0.f, 0.f, 0.f);
}

__global__ __launch_bounds__(256) void tv_splat_kernel(
    const float* __restrict__ center, const float* __restrict__ cov,
    const float* __restrict__ dens_r, const float* __restrict__ dens_i,
    const float* __restrict__ radius, const float* __restrict__ half_shape,
    float* __restrict__ out_r, float* __restrict__ out_i, int N)
{
    const int lane = threadIdx.x & 31;
    const int wave = threadIdx.x >> 5;
    const int g    = blockIdx.x * 8 + wave;
    if (g >= N) return;   // wave-uniform branch: EXEC stays all-ones inside

    // ---- per-Gaussian scalars -------------------------------------------
    const float cx = center[3 * g + 0];
    const float cy = center[3 * g + 1];
    const float cz = center[3 * g + 2];
    const float r  = radius[g];
    const float dr = dens_r[g];
    const float di = dens_i[g];
    const float ihx = 1.0f / half_shape[0];
    const float ihy = 1.0f / half_shape[1];
    const float ihz = 1.0f / half_shape[2];

    const int bx = (int)floorf(cx);
    const int by = (int)floorf(cy);
    const int bz = (int)floorf(cz);

    // window ∩ bbox ∩ grid extent per axis (in_bbox & in_grid become implicit)
    int lo0 = max(-RWIN, (int)floorf(cx - r) - bx);
    int hi0 = min( RWIN, (int)ceilf (cx + r) - bx);
    int lo1 = max(-RWIN, (int)floorf(cy - r) - by);
    int hi1 = min( RWIN, (int)ceilf (cy + r) - by);
    int lo2 = max(-RWIN, (int)floorf(cz - r) - bz);
    int hi2 = min( RWIN, (int)ceilf (cz + r) - bz);
    lo0 = max(lo0, -bx);  hi0 = min(hi0, D_DIM - 1 - bx);
    lo1 = max(lo1, -by);  hi1 = min(hi1, H_DIM - 1 - by);
    lo2 = max(lo2, -bz);  hi2 = min(hi2, W_DIM - 1 - bz);

    const int n0 = hi0 - lo0 + 1;
    const int n1 = hi1 - lo1 + 1;
    const int n2 = hi2 - lo2 + 1;
    if (n0 <= 0 || n1 <= 0 || n2 <= 0) return;   // wave-uniform
    const int n12 = n1 * n2;
    const int T   = n0 * n12;

    // ---- A matrix: cov_inv padded to 16x4 (rows 0..2 live) ---------------
    // lanes 0-15 hold A[M=lane][K=0] in .x, A[M=lane][K=1] in .y
    // lanes 16-31 hold A[M=lane-16][K=2] in .x, K=3 (zero) in .y
    const float* Cg = cov + 9 * g;
    v2f amat; amat.x = 0.0f; amat.y = 0.0f;
    if (lane < 3) {
        amat.x = Cg[lane * 3 + 0];
        amat.y = Cg[lane * 3 + 1];
    } else if (lane >= 16 && lane < 19) {
        amat.x = Cg[(lane - 16) * 3 + 2];
    }

    const int sub = lane & 15;

    // ---- 16 voxels per wave-iteration via V_WMMA_F32_16X16X4_F32 ---------
    for (int kbase = 0; kbase < T; kbase += 16) {
        const int  i     = kbase + sub;
        const bool valid = (i < T);
        const int  ii    = valid ? i : 0;

        int od = ii / n12;
        int rem = ii - od * n12;
        int oh = rem / n2;
        int ow = rem - oh * n2;
        od += lo0; oh += lo1; ow += lo2;

        const float dx = ((float)(bx + od) - cx) * ihx;
        const float dy = ((float)(by + oh) - cy) * ihy;
        const float dz = ((float)(bz + ow) - cz) * ihz;

        // B matrix: diff^T padded to 4x16
        // VGPR0: lanes 0-15 = K=0 row (dx), lanes 16-31 = K=2 row (dz)
        // VGPR1: lanes 0-15 = K=1 row (dy), lanes 16-31 = K=3 row (0)
        v2f bmat;
        bmat.x = (lane < 16) ? dx : dz;
        bmat.y = (lane < 16) ? dy : 0.0f;

        v8f dmat = {};
        // D = A(16x4) * B(4x16): lane n<16 gets (C·diff_n)[row] in dmat[row]
        dmat = __builtin_amdgcn_wmma_f32_16x16x4_f32(
            /*neg_a=*/false, amat, /*neg_b=*/false, bmat,
            /*c_mod=*/(short)0, dmat, /*reuse_a=*/false, /*reuse_b=*/false);

        const float Md = dx * dmat[0] + dy * dmat[1] + dz * dmat[2];

        const bool active = valid && (lane < 16) && (Md <= 9.0f);
        if (active) {
            // e^{-Md/2} = 2^{-Md * log2(e)/2}
            const float w = __builtin_amdgcn_exp2f(-0.72134752044448170f * Md);
            const int idx = ((bx + od) * H_DIM + (by + oh)) * W_DIM + (bz + ow);
            atomicAddF32(out_r + idx, w * dr);
            atomicAddF32(out_i + idx, w * di);
        }
        // reconverged here: EXEC all-ones again before next WMMA
    }
}

extern "C" void kernel_launch(void* const* d_in, const int* in_sizes, int n_in,
                              void* d_out, int out_size, void* d_ws, size_t ws_size,
                              hipStream_t stream) {
    (void)n_in; (void)d_ws; (void)ws_size; (void)out_size;

    const float* center     = (const float*)d_in[0];
    const float* cov        = (const float*)d_in[1];
    const float* dens_r     = (const float*)d_in[2];
    const float* dens_i     = (const float*)d_in[3];
    const float* radius     = (const float*)d_in[4];
    const float* half_shape = (const float*)d_in[5];

    const int N = in_sizes[0] / 3;
    const long long V = (long long)D_DIM * H_DIM * W_DIM;

    float* out_r = (float*)d_out;
    float* out_i = out_r + V;

    // zero both output volumes (harness poisons d_out with 0xAA)
    const int n4 = (int)(2 * V / 4);
    tv_zero_kernel<<<(n4 + 255) / 256, 256, 0, stream>>>((float4*)d_out, n4);

    // one wave32 per Gaussian, 8 waves per block
    const int blocks = (N + 7) / 8;
    tv_splat_kernel<<<blocks, 256, 0, stream>>>(
        center, cov, dens_r, dens_i, radius, half_shape, out_r, out_i, N);
}